// RelationalGraphConvLayer_66408784331250
// MI455X (gfx1250) — compile-verified
//
#include <hip/hip_runtime.h>

#define N_NODES 100000
#define R 8
#define E_PER_REL 80000
#define D 128           // D_IN == D_OUT
#define N_BASES 30
#define EPS 1e-5f

typedef float __attribute__((ext_vector_type(2))) v2f;
typedef float __attribute__((ext_vector_type(8))) v8f;

// ---------------------------------------------------------------------------
// Kernel 1: basis combine  W[r] = sum_b w_rel[r,b] * w_bases[b]  -> [R, D, D]
// ---------------------------------------------------------------------------
__global__ void rgcn_basis_kernel(const float* __restrict__ w_rel,
                                  const float* __restrict__ w_bases,
                                  float* __restrict__ W) {
    int idx = blockIdx.x * blockDim.x + threadIdx.x;   // R*D*D = 131072
    if (idx >= R * D * D) return;
    int r  = idx / (D * D);
    int io = idx % (D * D);
    float acc = 0.0f;
    #pragma unroll
    for (int b = 0; b < N_BASES; ++b)
        acc += w_rel[r * N_BASES + b] * w_bases[b * D * D + io];
    W[idx] = acc;
}

// ---------------------------------------------------------------------------
// Kernel 2: per-relation scatter aggregation.  One wave32 per edge; each lane
// gathers a float4 of X[dst] (L2-resident) and atomically adds into agg[src].
// ---------------------------------------------------------------------------
__global__ void rgcn_agg_kernel(const int* __restrict__ src,
                                const int* __restrict__ dst,
                                const float* __restrict__ X,
                                float* __restrict__ agg,
                                float* __restrict__ deg) {
    int t = blockIdx.x * blockDim.x + threadIdx.x;
    int e    = t >> 5;
    int lane = t & 31;
    if (e >= E_PER_REL) return;
    int s = src[e];
    int d = dst[e];
    const float4 v = *(const float4*)(X + (size_t)d * D + lane * 4);
    float* a = agg + (size_t)s * D + lane * 4;
    atomicAdd(a + 0, v.x);
    atomicAdd(a + 1, v.y);
    atomicAdd(a + 2, v.z);
    atomicAdd(a + 3, v.w);
    if (lane == 0) atomicAdd(deg + s, 1.0f);
}

// ---------------------------------------------------------------------------
// Kernel 3: out[n,:] += (agg[n,:] * 1/(deg[n]+eps)) @ Wr   via fp32 WMMA.
// Block = 256 threads = 8 waves; block covers 128 rows x 128 cols.
// Wave w covers rows [blockBase + 16w, +16) x all 128 cols (8 N-tiles).
//
// LDS layout for W is K-quad interleaved:  lsW[(k>>2)*512 + n*4 + (k&3)]
// so each lane's B fragment {W[k0][n], W[k0+1][n]} (k0 = kk + 2*kp, even) is
// two contiguous floats -> one ds_load_b64, and the 32 lanes of a wave hit
// all 64 LDS banks exactly once (conflict-free, no padding, 64 KB total).
// ---------------------------------------------------------------------------
__global__ void __launch_bounds__(256)
rgcn_gemm_kernel(const float* __restrict__ agg,
                 const float* __restrict__ deg,
                 const float* __restrict__ Wr,    // [D,D] row-major (k,n)
                 float* __restrict__ out) {
    __shared__ float lsW[D * D];                  // 64 KB, swizzled
    for (int idx = threadIdx.x; idx < D * D; idx += 256) {
        int k = idx >> 7;          // 0..127
        int n = idx & 127;
        lsW[(k >> 2) * 512 + n * 4 + (k & 3)] = Wr[idx];
    }
    __syncthreads();

    const int lane = threadIdx.x & 31;
    const int wave = threadIdx.x >> 5;            // 0..7
    const int m    = lane & 15;                   // row/col within 16
    const int kp   = lane >> 4;                   // K pair select (0 -> K0/K1, 1 -> K2/K3)
    const int rowBase = blockIdx.x * 128 + wave * 16;
    const int row     = rowBase + m;

    float rdeg = 0.0f;
    if (row < N_NODES) rdeg = 1.0f / (deg[row] + EPS);
    const float* arow = agg + (size_t)(row < N_NODES ? row : 0) * D;

    // per-lane base into swizzled W: column (n) and K-pair parity are fixed
    const float* lsWbase = lsW + m * 4 + 2 * kp;

    v8f c[8] = {};                                 // 8 C tiles (N = 0..127)

    for (int kk = 0; kk < D; kk += 4) {
        // A fragment (16x4 f32): lanes 0-15 hold K=kk,kk+1; lanes 16-31 K=kk+2,kk+3
        v2f a = *(const v2f*)(arow + kk + 2 * kp);
        a.x *= rdeg;
        a.y *= rdeg;
        const float* bq = lsWbase + (kk >> 2) * 512;
        #pragma unroll
        for (int nt = 0; nt < 8; ++nt) {
            v2f b = *(const v2f*)(bq + nt * 64);   // single ds_load_b64, no repacking
            c[nt] = __builtin_amdgcn_wmma_f32_16x16x4_f32(
                false, a, false, b, (short)0, c[nt], false, false);
        }
    }

    // C/D layout: VGPR g -> lanes 0-15: M=g, lanes 16-31: M=g+8 ; N = lane&15
    const int outRow0 = rowBase + kp * 8;
    #pragma unroll
    for (int g = 0; g < 8; ++g) {
        const int orow = outRow0 + g;
        if (orow < N_NODES) {
            #pragma unroll
            for (int nt = 0; nt < 8; ++nt)
                out[(size_t)orow * D + nt * 16 + m] += c[nt][g];
        }
    }
}

// ---------------------------------------------------------------------------
extern "C" void kernel_launch(void* const* d_in, const int* in_sizes, int n_in,
                              void* d_out, int out_size, void* d_ws, size_t ws_size,
                              hipStream_t stream) {
    const float* X        = (const float*)d_in[0];
    const int*   edge_src = (const int*)d_in[1];
    const int*   edge_dst = (const int*)d_in[2];
    const float* w_bases  = (const float*)d_in[3];
    const float* w_rel    = (const float*)d_in[4];
    // d_in[5] = l (branch selector, always 1 here)
    float* out = (float*)d_out;

    // Workspace layout: W[R*D*D] | agg[N*D] | deg[N]   (~52.2 MB total)
    float* W   = (float*)d_ws;
    float* agg = W + (size_t)R * D * D;
    float* deg = agg + (size_t)N_NODES * D;

    hipMemsetAsync(out, 0, (size_t)N_NODES * D * sizeof(float), stream);

    rgcn_basis_kernel<<<(R * D * D + 255) / 256, 256, 0, stream>>>(w_rel, w_bases, W);

    for (int r = 0; r < R; ++r) {
        // zero agg + deg (contiguous)
        hipMemsetAsync(agg, 0, ((size_t)N_NODES * D + N_NODES) * sizeof(float), stream);
        rgcn_agg_kernel<<<(E_PER_REL * 32 + 255) / 256, 256, 0, stream>>>(
            edge_src + (size_t)r * E_PER_REL,
            edge_dst + (size_t)r * E_PER_REL,
            X, agg, deg);
        rgcn_gemm_kernel<<<(N_NODES + 127) / 128, 256, 0, stream>>>(
            agg, deg, W + (size_t)r * D * D, out);
    }
}